// MaskedGCN_73942156968323
// MI455X (gfx1250) — compile-verified
//
#include <hip/hip_runtime.h>
#include <hip/hip_bf16.h>

typedef __attribute__((ext_vector_type(2))) float v2f;
typedef __attribute__((ext_vector_type(8))) float v8f;

// ---------------------------------------------------------------------------
// Workspace layout (floats unless noted):
//   cnt  : N u32          (zeroed each call)
//   agg1 : N*64           (zeroed each call; finalize overwrites with out1)
//   agg2 : N*40           (zeroed each call)
//   dinv : N
//   h1   : N*64
//   h2   : N*40
// ---------------------------------------------------------------------------

__global__ void gcn_zero_kernel(uint4* __restrict__ p, long n4) {
    long i = (long)blockIdx.x * blockDim.x + threadIdx.x;
    if (i < n4) p[i] = make_uint4(0u, 0u, 0u, 0u);
}

__global__ void gcn_deg_kernel(const long long* __restrict__ ei,
                               unsigned* __restrict__ cnt, long E) {
    long e = (long)blockIdx.x * blockDim.x + threadIdx.x;
    if (e < E) atomicAdd(&cnt[(long)ei[E + e]], 1u);
}

__global__ void gcn_dinv_kernel(const unsigned* __restrict__ cnt,
                                float* __restrict__ dinv, long N) {
    long i = (long)blockIdx.x * blockDim.x + threadIdx.x;
    if (i < N) dinv[i] = rsqrtf(1.0f + (float)cnt[i]);
}

// ---------------------------------------------------------------------------
// GEMM1: h1[N,64] = x[N,128] @ W1[128,64]   (no bias; bias added in finalize)
// One wave per 16-row tile. W1 staged in LDS in a fragment-interleaved layout:
//   Wl[(k>>1)*128 + 2*n + (k&1)] = W[k][n]
// so each B fragment {W[ka][n], W[ka+1][n]} is one aligned v2f -> ds_load_b64
// straight into an even VGPR pair (no repack movs before v_wmma).
// ---------------------------------------------------------------------------
__global__ void __launch_bounds__(256)
gcn_gemm1_kernel(const float* __restrict__ x, const float* __restrict__ W,
                 float* __restrict__ h, long N) {
    __shared__ float Wl[128 * 64];  // 32 KB
    for (int i = threadIdx.x; i < 128 * 64; i += 256) {
        int kk = i >> 6, nn = i & 63;
        Wl[(kk >> 1) * 128 + (nn << 1) + (kk & 1)] = W[i];
    }
    __syncthreads();

    long mtiles = (N + 15) >> 4;
    long tile = (long)blockIdx.x * 8 + (threadIdx.x >> 5);
    if (tile >= mtiles) return;

    int lane = threadIdx.x & 31;
    int row  = lane & 15;         // A: M index
    int kh   = lane >> 4;         // A/B: K half select
    int n    = lane & 15;         // B/C: N index within 16-col tile
    long m0  = tile << 4;

    long r = m0 + row; if (r >= N) r = N - 1;
    const float* xr = x + r * 128;
    __builtin_prefetch(xr, 0, 3);

    v8f c0 = {}, c1 = {}, c2 = {}, c3 = {};
    for (int k0 = 0; k0 < 128; k0 += 4) {
        int ka = k0 + 2 * kh;
        v2f a = *(const v2f*)(xr + ka);                 // global_load_b64
        const float* bp = Wl + (k0 / 2 + kh) * 128 + 2 * n;
        v2f b0 = *(const v2f*)(bp);                     // ds_load_b64 each
        v2f b1 = *(const v2f*)(bp + 32);
        v2f b2 = *(const v2f*)(bp + 64);
        v2f b3 = *(const v2f*)(bp + 96);
        c0 = __builtin_amdgcn_wmma_f32_16x16x4_f32(false, a, false, b0, (short)0, c0, false, false);
        c1 = __builtin_amdgcn_wmma_f32_16x16x4_f32(false, a, false, b1, (short)0, c1, false, false);
        c2 = __builtin_amdgcn_wmma_f32_16x16x4_f32(false, a, false, b2, (short)0, c2, false, false);
        c3 = __builtin_amdgcn_wmma_f32_16x16x4_f32(false, a, false, b3, (short)0, c3, false, false);
    }
    for (int rr = 0; rr < 8; ++rr) {
        long orow = m0 + rr + 8 * kh;          // C layout: lanes16-31 hold M=r+8
        if (orow >= N) continue;
        float* o = h + orow * 64 + n;
        o[0]  = c0[rr];
        o[16] = c1[rr];
        o[32] = c2[rr];
        o[48] = c3[rr];
    }
}

// ---------------------------------------------------------------------------
// GEMM2: h2[N,40] = out1[N,64] @ W2[64,40]  (cols padded to 48 with zeros)
// Same interleaved LDS layout, row-pair stride 96 floats (8B-aligned pairs).
// ---------------------------------------------------------------------------
__global__ void __launch_bounds__(256)
gcn_gemm2_kernel(const float* __restrict__ x, const float* __restrict__ W,
                 float* __restrict__ h, long N) {
    __shared__ float Wl[32 * 96];  // 12 KB: 32 k-pairs x 48 cols x 2
    for (int i = threadIdx.x; i < 64 * 48; i += 256) {
        int kk = i / 48, nn = i % 48;
        float v = (nn < 40) ? W[kk * 40 + nn] : 0.0f;
        Wl[(kk >> 1) * 96 + (nn << 1) + (kk & 1)] = v;
    }
    __syncthreads();

    long mtiles = (N + 15) >> 4;
    long tile = (long)blockIdx.x * 8 + (threadIdx.x >> 5);
    if (tile >= mtiles) return;

    int lane = threadIdx.x & 31;
    int row  = lane & 15;
    int kh   = lane >> 4;
    int n    = lane & 15;
    long m0  = tile << 4;

    long r = m0 + row; if (r >= N) r = N - 1;
    const float* xr = x + r * 64;
    __builtin_prefetch(xr, 0, 3);

    v8f c0 = {}, c1 = {}, c2 = {};
    for (int k0 = 0; k0 < 64; k0 += 4) {
        int ka = k0 + 2 * kh;
        v2f a = *(const v2f*)(xr + ka);
        const float* bp = Wl + (k0 / 2 + kh) * 96 + 2 * n;
        v2f b0 = *(const v2f*)(bp);
        v2f b1 = *(const v2f*)(bp + 32);
        v2f b2 = *(const v2f*)(bp + 64);
        c0 = __builtin_amdgcn_wmma_f32_16x16x4_f32(false, a, false, b0, (short)0, c0, false, false);
        c1 = __builtin_amdgcn_wmma_f32_16x16x4_f32(false, a, false, b1, (short)0, c1, false, false);
        c2 = __builtin_amdgcn_wmma_f32_16x16x4_f32(false, a, false, b2, (short)0, c2, false, false);
    }
    for (int rr = 0; rr < 8; ++rr) {
        long orow = m0 + rr + 8 * kh;
        if (orow >= N) continue;
        float* o = h + orow * 40 + n;
        o[0]  = c0[rr];
        o[16] = c1[rr];
        if (n < 8) o[32] = c2[rr];             // cols 32..39 only
    }
}

// ---------------------------------------------------------------------------
// Edge scatter: 64 consecutive threads per edge (wave-uniform src/dst/norm,
// contiguous per-edge feature atomics). F = 64 or 40 (f>=F lanes idle).
// The full h/agg working set fits in the 192MB L2, so this phase is
// L2-atomic-throughput bound; contiguity keeps atomics cacheline-local.
// ---------------------------------------------------------------------------
__global__ void gcn_scatter_kernel(const float* __restrict__ h,
                                   const long long* __restrict__ ei,
                                   const float* __restrict__ dinv,
                                   float* __restrict__ agg, long E, int F) {
    long g = (long)blockIdx.x * blockDim.x + threadIdx.x;
    long e = g >> 6;
    int  f = (int)(g & 63);
    if (e >= E || f >= F) return;
    long s = (long)ei[e];
    long d = (long)ei[E + e];
    float nrm = dinv[s] * dinv[d];
    atomicAdd(&agg[d * (long)F + f], h[s * (long)F + f] * nrm);
}

// agg1 <- relu(agg1 + h1 * dinv^2 + b1)   (layer-1 output, in place)
__global__ void gcn_finalize1_kernel(float* __restrict__ agg,
                                     const float* __restrict__ h,
                                     const float* __restrict__ dinv,
                                     const float* __restrict__ b, long total) {
    long i = (long)blockIdx.x * blockDim.x + threadIdx.x;
    if (i >= total) return;
    long row = i >> 6;
    int  col = (int)(i & 63);
    float dv = dinv[row];
    float v = agg[i] + h[i] * dv * dv + b[col];
    agg[i] = v > 0.0f ? v : 0.0f;
}

// One wave per node: finalize layer-2 + log_softmax over 40 classes.
__global__ void gcn_lsm_kernel(const float* __restrict__ agg,
                               const float* __restrict__ h,
                               const float* __restrict__ dinv,
                               const float* __restrict__ b,
                               float* __restrict__ out, long N) {
    int lane = threadIdx.x & 31;
    long node = (long)blockIdx.x * (blockDim.x >> 5) + (threadIdx.x >> 5);
    if (node >= N) return;
    float dv = dinv[node];
    float dd = dv * dv;
    const float* a = agg + node * 40;
    const float* hh = h + node * 40;
    float v0 = a[lane] + hh[lane] * dd + b[lane];
    float v1 = -INFINITY;
    if (lane < 8) v1 = a[lane + 32] + hh[lane + 32] * dd + b[lane + 32];
    float m = fmaxf(v0, v1);
    for (int o = 16; o >= 1; o >>= 1) m = fmaxf(m, __shfl_xor(m, o, 32));
    float s = __expf(v0 - m) + (lane < 8 ? __expf(v1 - m) : 0.0f);
    for (int o = 16; o >= 1; o >>= 1) s += __shfl_xor(s, o, 32);
    float lse = __logf(s) + m;
    out[node * 40 + lane] = v0 - lse;
    if (lane < 8) out[node * 40 + lane + 32] = v1 - lse;
}

extern "C" void kernel_launch(void* const* d_in, const int* in_sizes, int n_in,
                              void* d_out, int out_size, void* d_ws, size_t ws_size,
                              hipStream_t stream) {
    const float*     x  = (const float*)d_in[0];
    const long long* ei = (const long long*)d_in[1];   // [2, E] int64
    const float*     W1 = (const float*)d_in[2];
    const float*     b1 = (const float*)d_in[3];
    const float*     W2 = (const float*)d_in[4];
    const float*     b2 = (const float*)d_in[5];
    float* out = (float*)d_out;

    const long N = (long)in_sizes[0] / 128;
    const long E = (long)in_sizes[1] / 2;

    char* ws = (char*)d_ws;
    unsigned* cnt  = (unsigned*)ws;
    float*    agg1 = (float*)(ws + N * 4);
    float*    agg2 = agg1 + N * 64;
    float*    dinv = agg2 + N * 40;
    float*    h1   = dinv + N;
    float*    h2   = h1 + N * 64;

    const int B = 256;

    // Zero cnt + agg1 + agg2 (contiguous: N*(1+64+40) dwords).
    long zdw = N * 105;
    long z4  = (zdw + 3) / 4;   // may spill a few dwords into dinv (rewritten below)
    gcn_zero_kernel<<<(unsigned)((z4 + B - 1) / B), B, 0, stream>>>((uint4*)ws, z4);

    // Degrees + symmetric normalization.
    gcn_deg_kernel<<<(unsigned)((E + B - 1) / B), B, 0, stream>>>(ei, cnt, E);
    gcn_dinv_kernel<<<(unsigned)((N + B - 1) / B), B, 0, stream>>>(cnt, dinv, N);

    // Layer 1.
    long mtiles = (N + 15) / 16;
    unsigned gblocks = (unsigned)((mtiles + 7) / 8);
    gcn_gemm1_kernel<<<gblocks, B, 0, stream>>>(x, W1, h1, N);
    long t1 = E * 64;
    gcn_scatter_kernel<<<(unsigned)((t1 + B - 1) / B), B, 0, stream>>>(h1, ei, dinv, agg1, E, 64);
    long f1 = N * 64;
    gcn_finalize1_kernel<<<(unsigned)((f1 + B - 1) / B), B, 0, stream>>>(agg1, h1, dinv, b1, f1);

    // Layer 2 (input = agg1 holding relu'd layer-1 output).
    gcn_gemm2_kernel<<<gblocks, B, 0, stream>>>(agg1, W2, h2, N);
    gcn_scatter_kernel<<<(unsigned)((t1 + B - 1) / B), B, 0, stream>>>(h2, ei, dinv, agg2, E, 40);

    // Finalize + log_softmax (one wave32 per node).
    unsigned lblocks = (unsigned)((N + 7) / 8);
    gcn_lsm_kernel<<<lblocks, B, 0, stream>>>(agg2, h2, dinv, b2, out, N);
}